// DynamicGCNLayer_47330539602429
// MI455X (gfx1250) — compile-verified
//
#include <hip/hip_runtime.h>
#include <math.h>

#define FD 128            // feature dim
#define LXS 132           // LDS row stride (pad 4 floats -> conflict-free banks)

typedef __attribute__((ext_vector_type(2))) float v2f;
typedef __attribute__((ext_vector_type(8))) float v8f;

// ---------------- degree / normalization ----------------

__global__ void init_deg_kernel(float* deg, int n) {
    int i = blockIdx.x * blockDim.x + threadIdx.x;
    if (i < n) deg[i] = 1.0f;               // self loop counted
}

__global__ void scatter_deg_kernel(const long long* __restrict__ dst, float* deg, long long e) {
    long long i = (long long)blockIdx.x * blockDim.x + threadIdx.x;
    if (i < e) atomicAdd(&deg[(int)dst[i]], 1.0f);
}

__global__ void deg_to_dinv_kernel(float* deg, int n) {
    int i = blockIdx.x * blockDim.x + threadIdx.x;
    if (i < n) deg[i] = rsqrtf(deg[i]);     // deg >= 1 always (self loop)
}

// ---------------- xw = x @ W  (f32 WMMA, 32 rows/block, 2 chains/wave) ----

__global__ __launch_bounds__(256) void gemm_xw_kernel(
    const float* __restrict__ x, const float* __restrict__ W,
    float* __restrict__ xw, int n)
{
    __shared__ float lx[32 * LXS];
    int row0 = blockIdx.x * 32;

    // stage 32x128 tile with b128 loads
    for (int t = threadIdx.x; t < 32 * (FD / 4); t += 256) {
        int r  = t >> 5;            // 32 float4 per row
        int c4 = (t & 31) * 4;
        int row = row0 + r; if (row >= n) row = n - 1;
        float4 v = *(const float4*)(x + (size_t)row * FD + c4);
        float* p = &lx[r * LXS + c4];
        p[0] = v.x; p[1] = v.y; p[2] = v.z; p[3] = v.w;
    }
    __syncthreads();

    int lane = threadIdx.x & 31;
    int wv   = threadIdx.x >> 5;        // 8 waves cover 8 column tiles
    int m    = lane & 15;
    int half = lane >> 4;
    int col  = wv * 16 + m;             // B/C/D column index for this lane

    v8f acc0 = {}, acc1 = {};
    #pragma unroll
    for (int kk = 0; kk < FD; kk += 4) {
        int kb = kk + half * 2;
        v2f b, a0, a1;
        b.x  = W[(size_t)kb       * FD + col];
        b.y  = W[(size_t)(kb + 1) * FD + col];
        a0.x = lx[m * LXS + kb];        a0.y = lx[m * LXS + kb + 1];
        a1.x = lx[(m + 16) * LXS + kb]; a1.y = lx[(m + 16) * LXS + kb + 1];
        acc0 = __builtin_amdgcn_wmma_f32_16x16x4_f32(false, a0, false, b, (short)0, acc0, false, false);
        acc1 = __builtin_amdgcn_wmma_f32_16x16x4_f32(false, a1, false, b, (short)0, acc1, false, false);
    }

    #pragma unroll
    for (int r = 0; r < 8; ++r) {
        int rowA = row0 + half * 8 + r;
        int rowB = rowA + 16;
        if (rowA < n) xw[(size_t)rowA * FD + col] = acc0[r];
        if (rowB < n) xw[(size_t)rowB * FD + col] = acc1[r];
    }
}

// ---- agg init: self-loop contribution + bias folded in (float4) ----

__global__ void init_agg_kernel(const float* __restrict__ xw, const float* __restrict__ dinv,
                                const float* __restrict__ b, float* agg, long long total4)
{
    long long i = (long long)blockIdx.x * blockDim.x + threadIdx.x;
    if (i < total4) {
        int node = (int)(i >> 5);
        int c4   = (int)(i & 31) * 4;
        float dv = dinv[node]; dv *= dv;
        float4 v  = *(const float4*)(xw + ((size_t)node * FD + c4));
        float4 bb = *(const float4*)(b + c4);
        float4 o;
        o.x = v.x * dv + bb.x; o.y = v.y * dv + bb.y;
        o.z = v.z * dv + bb.z; o.w = v.w * dv + bb.w;
        *(float4*)(agg + ((size_t)node * FD + c4)) = o;
    }
}

// ---- edge scatter: agg[dst] += xw[src] * dinv[src]*dinv[dst] (b128 gather) ----

__global__ void scatter_edges_kernel(const long long* __restrict__ src,
                                     const long long* __restrict__ dst,
                                     const float* __restrict__ xw,
                                     const float* __restrict__ dinv,
                                     float* agg, long long e)
{
    long long idx  = (long long)blockIdx.x * blockDim.x + threadIdx.x;
    long long eidx = idx >> 5;
    if (eidx < e) {
        int c4 = (int)(idx & 31) * 4;
        int s  = (int)src[eidx];
        int d  = (int)dst[eidx];
        float nrm = dinv[s] * dinv[d];
        float4 v  = *(const float4*)(xw + (size_t)s * FD + c4);
        float* ap = agg + (size_t)d * FD + c4;
        atomicAdd(ap + 0, v.x * nrm);
        atomicAdd(ap + 1, v.y * nrm);
        atomicAdd(ap + 2, v.z * nrm);
        atomicAdd(ap + 3, v.w * nrm);
    }
}

// ------- fused GRU cell (32 rows/block, 12 WMMA chains/wave) -------

__global__ __launch_bounds__(256) void gru_fused_kernel(
    const float* __restrict__ hcur,   // GCN output = GRU "input"
    const float* __restrict__ xprev,  // GRU hidden state
    const float* __restrict__ Wih, const float* __restrict__ Whh,
    const float* __restrict__ bih, const float* __restrict__ bhh,
    float* __restrict__ out, int n)
{
    __shared__ float lh[32 * LXS];
    __shared__ float lp[32 * LXS];
    int row0 = blockIdx.x * 32;

    for (int t = threadIdx.x; t < 32 * (FD / 4); t += 256) {
        int r  = t >> 5;
        int c4 = (t & 31) * 4;
        int row = row0 + r; if (row >= n) row = n - 1;
        float4 vh = *(const float4*)(hcur  + (size_t)row * FD + c4);
        float4 vp = *(const float4*)(xprev + (size_t)row * FD + c4);
        float* ph = &lh[r * LXS + c4];
        float* pp = &lp[r * LXS + c4];
        ph[0] = vh.x; ph[1] = vh.y; ph[2] = vh.z; ph[3] = vh.w;
        pp[0] = vp.x; pp[1] = vp.y; pp[2] = vp.z; pp[3] = vp.w;
    }
    __syncthreads();

    int lane = threadIdx.x & 31;
    int wv   = threadIdx.x >> 5;
    int m    = lane & 15;
    int half = lane >> 4;
    int col  = wv * 16 + m;

    // gi[n,j] = sum_k hcur[n,k] * Wih[j,k]  -> B[k][j] = Wih[j*FD + k]
    const float* wr = Wih + (size_t)(col          ) * FD;
    const float* wz = Wih + (size_t)(col +     FD ) * FD;
    const float* wn = Wih + (size_t)(col + 2 * FD ) * FD;
    const float* vr = Whh + (size_t)(col          ) * FD;
    const float* vz = Whh + (size_t)(col +     FD ) * FD;
    const float* vn = Whh + (size_t)(col + 2 * FD ) * FD;

    v8f ir0 = {}, iz0 = {}, in0 = {}, hr0 = {}, hz0 = {}, hn0 = {};
    v8f ir1 = {}, iz1 = {}, in1 = {}, hr1 = {}, hz1 = {}, hn1 = {};

    #pragma unroll 4
    for (int kk = 0; kk < FD; kk += 4) {
        int kb = kk + half * 2;
        v2f ai0, ai1, ah0, ah1, b0, b1, b2, b3, b4, b5;
        ai0.x = lh[m * LXS + kb];        ai0.y = lh[m * LXS + kb + 1];
        ai1.x = lh[(m + 16) * LXS + kb]; ai1.y = lh[(m + 16) * LXS + kb + 1];
        ah0.x = lp[m * LXS + kb];        ah0.y = lp[m * LXS + kb + 1];
        ah1.x = lp[(m + 16) * LXS + kb]; ah1.y = lp[(m + 16) * LXS + kb + 1];
        b0.x = wr[kb]; b0.y = wr[kb + 1];
        b1.x = wz[kb]; b1.y = wz[kb + 1];
        b2.x = wn[kb]; b2.y = wn[kb + 1];
        b3.x = vr[kb]; b3.y = vr[kb + 1];
        b4.x = vz[kb]; b4.y = vz[kb + 1];
        b5.x = vn[kb]; b5.y = vn[kb + 1];
        ir0 = __builtin_amdgcn_wmma_f32_16x16x4_f32(false, ai0, false, b0, (short)0, ir0, false, false);
        ir1 = __builtin_amdgcn_wmma_f32_16x16x4_f32(false, ai1, false, b0, (short)0, ir1, false, false);
        iz0 = __builtin_amdgcn_wmma_f32_16x16x4_f32(false, ai0, false, b1, (short)0, iz0, false, false);
        iz1 = __builtin_amdgcn_wmma_f32_16x16x4_f32(false, ai1, false, b1, (short)0, iz1, false, false);
        in0 = __builtin_amdgcn_wmma_f32_16x16x4_f32(false, ai0, false, b2, (short)0, in0, false, false);
        in1 = __builtin_amdgcn_wmma_f32_16x16x4_f32(false, ai1, false, b2, (short)0, in1, false, false);
        hr0 = __builtin_amdgcn_wmma_f32_16x16x4_f32(false, ah0, false, b3, (short)0, hr0, false, false);
        hr1 = __builtin_amdgcn_wmma_f32_16x16x4_f32(false, ah1, false, b3, (short)0, hr1, false, false);
        hz0 = __builtin_amdgcn_wmma_f32_16x16x4_f32(false, ah0, false, b4, (short)0, hz0, false, false);
        hz1 = __builtin_amdgcn_wmma_f32_16x16x4_f32(false, ah1, false, b4, (short)0, hz1, false, false);
        hn0 = __builtin_amdgcn_wmma_f32_16x16x4_f32(false, ah0, false, b5, (short)0, hn0, false, false);
        hn1 = __builtin_amdgcn_wmma_f32_16x16x4_f32(false, ah1, false, b5, (short)0, hn1, false, false);
    }

    float bir = bih[col], biz = bih[col + FD], bin = bih[col + 2 * FD];
    float bhr = bhh[col], bhz = bhh[col + FD], bhn = bhh[col + 2 * FD];

    #pragma unroll
    for (int r = 0; r < 8; ++r) {
        int lrow = r + half * 8;
        {   // tile 0
            int row = row0 + lrow;
            float h  = lp[lrow * LXS + col];
            float rg = 1.0f / (1.0f + __expf(-(ir0[r] + bir + hr0[r] + bhr)));
            float zg = 1.0f / (1.0f + __expf(-(iz0[r] + biz + hz0[r] + bhz)));
            float ng = tanhf(in0[r] + bin + rg * (hn0[r] + bhn));
            if (row < n) out[(size_t)row * FD + col] = (1.0f - zg) * ng + zg * h;
        }
        {   // tile 1
            int row = row0 + 16 + lrow;
            float h  = lp[(lrow + 16) * LXS + col];
            float rg = 1.0f / (1.0f + __expf(-(ir1[r] + bir + hr1[r] + bhr)));
            float zg = 1.0f / (1.0f + __expf(-(iz1[r] + biz + hz1[r] + bhz)));
            float ng = tanhf(in1[r] + bin + rg * (hn1[r] + bhn));
            if (row < n) out[(size_t)row * FD + col] = (1.0f - zg) * ng + zg * h;
        }
    }
}

// ---------------- launcher ----------------

extern "C" void kernel_launch(void* const* d_in, const int* in_sizes, int n_in,
                              void* d_out, int out_size, void* d_ws, size_t ws_size,
                              hipStream_t stream)
{
    const float*     x     = (const float*)d_in[0];
    const long long* ei    = (const long long*)d_in[1];   // int64 [2,E]
    const float*     xprev = (const float*)d_in[2];
    const float*     W     = (const float*)d_in[3];
    const float*     b     = (const float*)d_in[4];
    const float*     Wih   = (const float*)d_in[5];
    const float*     Whh   = (const float*)d_in[6];
    const float*     bih   = (const float*)d_in[7];
    const float*     bhh   = (const float*)d_in[8];
    float*           out   = (float*)d_out;

    int       n = in_sizes[0] / FD;
    long long e = (long long)in_sizes[1] / 2;
    const long long* src = ei;
    const long long* dst = ei + e;

    float* ws   = (float*)d_ws;
    float* dinv = ws;                           // n floats (padded slot)
    float* xw   = ws + 102400;                  // n*128 floats
    float* agg  = xw + (size_t)n * FD;          // n*128 floats

    int tiles32 = (n + 31) / 32;

    init_deg_kernel   <<<(n + 255) / 256, 256, 0, stream>>>(dinv, n);
    scatter_deg_kernel<<<(unsigned)((e + 255) / 256), 256, 0, stream>>>(dst, dinv, e);
    deg_to_dinv_kernel<<<(n + 255) / 256, 256, 0, stream>>>(dinv, n);

    gemm_xw_kernel<<<tiles32, 256, 0, stream>>>(x, W, xw, n);

    long long total4 = (long long)n * (FD / 4);
    init_agg_kernel<<<(unsigned)((total4 + 255) / 256), 256, 0, stream>>>(xw, dinv, b, agg, total4);

    long long work = e * (FD / 4);
    scatter_edges_kernel<<<(unsigned)((work + 255) / 256), 256, 0, stream>>>(src, dst, xw, dinv, agg, e);

    gru_fused_kernel<<<tiles32, 256, 0, stream>>>(agg, xprev, Wih, Whh, bih, bhh, out, n);
}